// SuperGAT_49323404427979
// MI455X (gfx1250) — compile-verified
//
#include <hip/hip_runtime.h>
#include <math.h>

typedef __attribute__((ext_vector_type(2))) float v2f;
typedef __attribute__((ext_vector_type(8))) float v8f;

#define NHEADS 4
#define CDIM 64
#define HC 256   /* NHEADS*CDIM */
#define NEG_SLOPE 0.2f

__device__ __forceinline__ float wave_sum(float v) {
#pragma unroll
  for (int off = 16; off >= 1; off >>= 1) v += __shfl_xor(v, off, 32);
  return v;
}
__device__ __forceinline__ float wave_max(float v) {
#pragma unroll
  for (int off = 16; off >= 1; off >>= 1) v = fmaxf(v, __shfl_xor(v, off, 32));
  return v;
}
// float atomic max via sign-aware integer atomics (m initialized to -inf)
__device__ __forceinline__ void atomic_max_f32(float* addr, float val) {
  if (val >= 0.0f) atomicMax((int*)addr, __float_as_int(val));
  else             atomicMin((unsigned int*)addr, __float_as_uint(val));
}

// C[M,Nc] = A[M,K] * B[K,Nc], fp32 exact, one 16x16 tile per wave via
// V_WMMA_F32_16X16X4_F32 (K stepped by 4).
__global__ void gemm_wmma_f32(const float* __restrict__ A,
                              const float* __restrict__ B,
                              float* __restrict__ Cc,
                              int M, int K, int Nc) {
  int wid = blockIdx.x * (blockDim.x >> 5) + (threadIdx.x >> 5);
  int ntn = Nc >> 4;
  int mt  = (M + 15) >> 4;
  if (wid >= mt * ntn) return;
  int tm = wid / ntn, tn = wid - tm * ntn;
  int lane = threadIdx.x & 31;
  int half = lane >> 4;        // 0: K-pair {0,1}, rows 0-7 of C ; 1: K-pair {2,3}, rows 8-15
  int l15  = lane & 15;
  int arow = tm * 16 + l15;
  if (arow >= M) arow = M - 1;               // clamp for (unused) tail; stores guarded
  const float* Ap = A + (size_t)arow * K + 2 * half;           // A[arow, k+2*half]
  const float* Bp = B + (size_t)(2 * half) * Nc + tn * 16 + l15; // B[k+2*half, n]
  v8f acc = {};
  for (int k = 0; k < K; k += 4) {
    v2f af, bf;
    af.x = Ap[k];                 af.y = Ap[k + 1];
    bf.x = Bp[(size_t)k * Nc];    bf.y = Bp[(size_t)(k + 1) * Nc];
    acc = __builtin_amdgcn_wmma_f32_16x16x4_f32(false, af, false, bf,
                                                (short)0, acc, false, false);
  }
  int crow0 = tm * 16 + 8 * half;
#pragma unroll
  for (int r = 0; r < 8; ++r) {
    int row = crow0 + r;
    if (row < M) Cc[(size_t)row * Nc + tn * 16 + l15] = acc[r];
  }
}

// nAl[n,h] = dot(h[n,h,:], al[h,:]);  nAr likewise with ar. One wave per (n,head).
__global__ void node_dots(const float* __restrict__ h,
                          const float* __restrict__ al,
                          const float* __restrict__ ar,
                          float* __restrict__ nAl, float* __restrict__ nAr,
                          int Nn) {
  int wid = blockIdx.x * (blockDim.x >> 5) + (threadIdx.x >> 5);
  if (wid >= Nn * NHEADS) return;
  int lane = threadIdx.x & 31;
  int n = wid >> 2, head = wid & 3;
  const float* hp = h + (size_t)n * HC + head * CDIM;
  float a0 = hp[lane] * al[head * CDIM + lane] +
             hp[lane + 32] * al[head * CDIM + lane + 32];
  float r0 = hp[lane] * ar[head * CDIM + lane] +
             hp[lane + 32] * ar[head * CDIM + lane + 32];
  a0 = wave_sum(a0);
  r0 = wave_sum(r0);
  if (lane == 0) { nAl[wid] = a0; nAr[wid] = r0; }
}

__global__ void init_bufs(float* __restrict__ acc, float* __restrict__ m,
                          float* __restrict__ s, int Nn) {
  int idx = blockIdx.x * blockDim.x + threadIdx.x;
  if (idx < Nn * HC) acc[idx] = 0.0f;
  if (idx < Nn * NHEADS) { m[idx] = -__builtin_inff(); s[idx] = 0.0f; }
}

// Pass 1: per-edge logits (dot over C per head), alpha = lrelu((al.xj+ar.xi)*sigmoid(logit)),
// atomic segment-max into m[dst,head]. One wave per edge; self-loops are edges >= E.
__global__ void edge_logits_alpha(const float* __restrict__ h,
                                  const float* __restrict__ nAl,
                                  const float* __restrict__ nAr,
                                  const long long* __restrict__ ei,
                                  float* __restrict__ alpha,
                                  float* __restrict__ mbuf,
                                  int E, int ET) {
  int e = blockIdx.x * (blockDim.x >> 5) + (threadIdx.x >> 5);
  if (e >= ET) return;
  int lane = threadIdx.x & 31;
  int src, dst;
  if (e < E) { src = (int)ei[e]; dst = (int)ei[E + e]; }
  else       { src = dst = e - E; }
  const float* hs = h + (size_t)src * HC;
  const float* hd = h + (size_t)dst * HC;
#pragma unroll
  for (int head = 0; head < NHEADS; ++head) {
    const float* a = hs + head * CDIM;
    const float* b = hd + head * CDIM;
    float p = a[lane] * b[lane] + a[lane + 32] * b[lane + 32];
    p = wave_sum(p);
    if (lane == 0) {
      float sg = 1.0f / (1.0f + expf(-p));
      float v = (nAl[src * NHEADS + head] + nAr[dst * NHEADS + head]) * sg;
      v = v > 0.0f ? v : NEG_SLOPE * v;
      alpha[(size_t)e * NHEADS + head] = v;
      atomic_max_f32(&mbuf[dst * NHEADS + head], v);
    }
  }
}

// Pass 2: e = exp(alpha - m[dst]); segment-sum into s[dst]. One thread per (edge,head).
__global__ void edge_exp_sum(const long long* __restrict__ ei,
                             float* __restrict__ alpha,
                             const float* __restrict__ mbuf,
                             float* __restrict__ sbuf,
                             int E, int ET) {
  int idx = blockIdx.x * blockDim.x + threadIdx.x;
  if (idx >= ET * NHEADS) return;
  int e = idx >> 2, head = idx & 3;
  int dst = (e < E) ? (int)ei[E + e] : e - E;
  float v = expf(alpha[idx] - mbuf[dst * NHEADS + head]);
  alpha[idx] = v;
  atomicAdd(&sbuf[dst * NHEADS + head], v);
}

// Pass 3: out[dst,h,c] += h[src,h,c] * e/(s[dst,h]+eps). One wave per edge, 8 elems/lane.
__global__ void edge_scatter(const float* __restrict__ h,
                             const long long* __restrict__ ei,
                             const float* __restrict__ alpha,
                             const float* __restrict__ sbuf,
                             float* __restrict__ acc,
                             int E, int ET) {
  int e = blockIdx.x * (blockDim.x >> 5) + (threadIdx.x >> 5);
  if (e >= ET) return;
  int lane = threadIdx.x & 31;
  int src, dst;
  if (e < E) { src = (int)ei[e]; dst = (int)ei[E + e]; }
  else       { src = dst = e - E; }
  float w[NHEADS];
#pragma unroll
  for (int head = 0; head < NHEADS; ++head)
    w[head] = alpha[(size_t)e * NHEADS + head] /
              (sbuf[dst * NHEADS + head] + 1e-16f);
  const float* hs = h + (size_t)src * HC;
  float* po = acc + (size_t)dst * HC;
#pragma unroll
  for (int i = 0; i < 8; ++i) {
    int j = lane + 32 * i;          // head = i>>1 (static under unroll)
    atomicAdd(&po[j], hs[j] * w[i >> 1]);
  }
}

// Layers 1-2: head-mean + bias + relu -> next layer input [N,64]
__global__ void finalize_relu(const float* __restrict__ acc,
                              const float* __restrict__ bias,
                              float* __restrict__ xout, int Nn) {
  int idx = blockIdx.x * blockDim.x + threadIdx.x;
  if (idx >= Nn * CDIM) return;
  int n = idx >> 6, c = idx & 63;
  const float* p = acc + (size_t)n * HC + c;
  float v = 0.25f * (p[0] + p[CDIM] + p[2 * CDIM] + p[3 * CDIM]) + bias[c];
  xout[idx] = v > 0.0f ? v : 0.0f;
}

// Layer 3: head-mean + bias + relu + log_softmax over 64 feats. One wave per node.
__global__ void finalize_lsm(const float* __restrict__ acc,
                             const float* __restrict__ bias,
                             float* __restrict__ out, int Nn) {
  int n = blockIdx.x * (blockDim.x >> 5) + (threadIdx.x >> 5);
  if (n >= Nn) return;
  int lane = threadIdx.x & 31;
  const float* p = acc + (size_t)n * HC;
  float v0 = 0.25f * (p[lane] + p[lane + CDIM] + p[lane + 2 * CDIM] +
                      p[lane + 3 * CDIM]) + bias[lane];
  float v1 = 0.25f * (p[lane + 32] + p[lane + 32 + CDIM] + p[lane + 32 + 2 * CDIM] +
                      p[lane + 32 + 3 * CDIM]) + bias[lane + 32];
  v0 = v0 > 0.0f ? v0 : 0.0f;
  v1 = v1 > 0.0f ? v1 : 0.0f;
  float mx = wave_max(fmaxf(v0, v1));
  float sm = wave_sum(expf(v0 - mx) + expf(v1 - mx));
  float ls = mx + logf(sm);
  out[(size_t)n * CDIM + lane] = v0 - ls;
  out[(size_t)n * CDIM + lane + 32] = v1 - ls;
}

extern "C" void kernel_launch(void* const* d_in, const int* in_sizes, int n_in,
                              void* d_out, int out_size, void* d_ws, size_t ws_size,
                              hipStream_t stream) {
  const float* x = (const float*)d_in[0];
  const long long* ei = (const long long*)d_in[1];
  int Nn = in_sizes[0] / 256;      // F_IN = 256
  int E  = in_sizes[1] / 2;
  int ET = E + Nn;                 // edges + self loops

  // workspace layout (floats); total ~130 MB
  float* ws   = (float*)d_ws;
  float* h    = ws;  ws += (size_t)Nn * HC;       // per-layer features [N,256]
  float* acc  = ws;  ws += (size_t)Nn * HC;       // aggregation accumulator
  float* nAl  = ws;  ws += (size_t)Nn * NHEADS;
  float* nAr  = ws;  ws += (size_t)Nn * NHEADS;
  float* mbuf = ws;  ws += (size_t)Nn * NHEADS;   // segment max
  float* sbuf = ws;  ws += (size_t)Nn * NHEADS;   // segment sum
  float* alph = ws;  ws += (size_t)ET * NHEADS;   // per-edge alpha / exp
  float* xcur = ws;  ws += (size_t)Nn * CDIM;     // layer 2/3 input

  const int TB = 256, WPB = TB / 32;
  for (int layer = 0; layer < 3; ++layer) {
    const float* X = (layer == 0) ? x : xcur;
    int K = (layer == 0) ? 256 : 64;
    const float* W  = (const float*)d_in[2 + layer * 4];
    const float* al = (const float*)d_in[3 + layer * 4];
    const float* ar = (const float*)d_in[4 + layer * 4];
    const float* bb = (const float*)d_in[5 + layer * 4];

    int gw = ((Nn + 15) / 16) * (HC / 16);
    gemm_wmma_f32<<<(gw + WPB - 1) / WPB, TB, 0, stream>>>(X, W, h, Nn, K, HC);
    node_dots<<<(Nn * NHEADS + WPB - 1) / WPB, TB, 0, stream>>>(h, al, ar, nAl, nAr, Nn);
    init_bufs<<<(Nn * HC + TB - 1) / TB, TB, 0, stream>>>(acc, mbuf, sbuf, Nn);
    edge_logits_alpha<<<(ET + WPB - 1) / WPB, TB, 0, stream>>>(h, nAl, nAr, ei, alph, mbuf, E, ET);
    edge_exp_sum<<<(ET * NHEADS + TB - 1) / TB, TB, 0, stream>>>(ei, alph, mbuf, sbuf, E, ET);
    edge_scatter<<<(ET + WPB - 1) / WPB, TB, 0, stream>>>(h, ei, alph, sbuf, acc, E, ET);
    if (layer < 2)
      finalize_relu<<<(Nn * CDIM + TB - 1) / TB, TB, 0, stream>>>(acc, bb, xcur, Nn);
    else
      finalize_lsm<<<(Nn + WPB - 1) / WPB, TB, 0, stream>>>(acc, bb, (float*)d_out, Nn);
  }
}